// DeepIRT_25847113187477
// MI455X (gfx1250) — compile-verified
//
#include <hip/hip_runtime.h>
#include <math.h>

// DeepIRT / DKVMN forward, MI455X (gfx1250, wave32).
//   B=128, L=500, D=64, M=50, NUM_C=10000.
// Kernel 1 (4000 waves): gathers + WMMA f32 GEMMs (k@Mk^T softmax, erase/add gates).
// Kernel 2 (128 blocks): serial 500-step memory scan; Mv column-resident in VGPRs,
//                        w[t] double-buffered in LDS via async global->LDS loads.
// Kernel 3 (4000 waves): WMMA f32 GEMM over concat(read,k) @ f_W + output head.

#define NUM_C 10000
#define DIM_S 64
#define SIZE_M 50
#define BB 128
#define LL 500
#define TOK (BB * LL)          // 64000 tokens, 4000 16-token tiles

typedef __attribute__((ext_vector_type(2))) float v2f;
typedef __attribute__((ext_vector_type(8))) float v8f;

#if __has_builtin(__builtin_amdgcn_global_load_async_to_lds_b32) && \
    __has_builtin(__builtin_amdgcn_s_wait_asynccnt)
#define HAVE_ASYNC_LDS 1
#else
#define HAVE_ASYNC_LDS 0
#endif

// ---------------------------------------------------------------------------
// Kernel 1: per 16-token tile (one wave32).
//   w = softmax(k @ Mk^T)   [16 x 50]
//   e = sigmoid(v @ e_W+b)  [16 x 64]
//   a = tanh   (v @ a_W+b)  [16 x 64]
// All GEMMs: V_WMMA_F32_16X16X4_F32, K-loop of 16 (K=64).
// ---------------------------------------------------------------------------
__global__ __launch_bounds__(32) void k_gates(
    const int* __restrict__ q, const int* __restrict__ r,
    const float* __restrict__ k_emb, const float* __restrict__ v_emb,
    const float* __restrict__ Mk,
    const float* __restrict__ eW, const float* __restrict__ eb,
    const float* __restrict__ aW, const float* __restrict__ abv,
    float* __restrict__ k_s, float* __restrict__ w_s,
    float* __restrict__ e_s, float* __restrict__ a_s)
{
    __shared__ float sK[16][65];   // padded stride: bank-conflict free
    __shared__ float sV[16][65];
    __shared__ float sW[16][65];   // w logits (cols >= 50 are exactly 0)

    const int lane = threadIdx.x;            // 0..31
    const int tok0 = blockIdx.x * 16;
    const int row  = lane & 15;              // A row within tile
    const int koff = (lane >> 4) * 2;        // f32 A/B frag K sub-offset
    const int col0 = lane & 15;              // B/C column within n-tile

    // Gather k/v rows (64 floats each) into LDS; mirror k to workspace.
    for (int i = 0; i < 16; ++i) {
        const int tok = tok0 + i;
        const int qi  = q[tok];
        const int xi  = qi + NUM_C * r[tok];
        const float k0 = k_emb[qi * DIM_S + lane];
        const float k1 = k_emb[qi * DIM_S + lane + 32];
        sK[i][lane] = k0;  sK[i][lane + 32] = k1;
        sV[i][lane]      = v_emb[xi * DIM_S + lane];
        sV[i][lane + 32] = v_emb[xi * DIM_S + lane + 32];
        k_s[tok * DIM_S + lane]      = k0;
        k_s[tok * DIM_S + lane + 32] = k1;
    }
    __syncthreads();

    // ---- logits = k @ Mk^T  (n padded 50 -> 64 with zero B-frags) ----
    for (int nt = 0; nt < 4; ++nt) {
        const int n = nt * 16 + col0;
        v8f acc = {};
#pragma unroll
        for (int kk = 0; kk < 16; ++kk) {
            const int kb = kk * 4 + koff;
            v2f a; a.x = sK[row][kb]; a.y = sK[row][kb + 1];
            v2f b;
            b.x = (n < SIZE_M) ? Mk[n * DIM_S + kb]     : 0.0f;
            b.y = (n < SIZE_M) ? Mk[n * DIM_S + kb + 1] : 0.0f;
            acc = __builtin_amdgcn_wmma_f32_16x16x4_f32(
                false, a, false, b, (short)0, acc, false, false);
        }
#pragma unroll
        for (int j = 0; j < 8; ++j) {
            const int m = j + ((lane >> 4) << 3);
            sW[m][nt * 16 + col0] = acc[j];
        }
    }
    __syncthreads();

    // ---- softmax over 50 slots; lanes 0..15 own one token-row each ----
    if (lane < 16) {
        float mx = -3.4e38f;
#pragma unroll
        for (int m = 0; m < SIZE_M; ++m) mx = fmaxf(mx, sW[lane][m]);
        float ex[SIZE_M];
        float sum = 0.0f;
#pragma unroll
        for (int m = 0; m < SIZE_M; ++m) { ex[m] = expf(sW[lane][m] - mx); sum += ex[m]; }
        const float inv = 1.0f / sum;
#pragma unroll
        for (int m = 0; m < SIZE_M; ++m)
            w_s[(tok0 + lane) * SIZE_M + m] = ex[m] * inv;
    }

    // ---- e = sigmoid(v@e_W + e_b), a = tanh(v@a_W + a_b) ----
    for (int nt = 0; nt < 4; ++nt) {
        const int n = nt * 16 + col0;
        v8f accE = {}, accA = {};
#pragma unroll
        for (int kk = 0; kk < 16; ++kk) {
            const int kb = kk * 4 + koff;
            v2f a;  a.x  = sV[row][kb];          a.y  = sV[row][kb + 1];
            v2f be; be.x = eW[kb * DIM_S + n];   be.y = eW[(kb + 1) * DIM_S + n];
            v2f ba; ba.x = aW[kb * DIM_S + n];   ba.y = aW[(kb + 1) * DIM_S + n];
            accE = __builtin_amdgcn_wmma_f32_16x16x4_f32(
                false, a, false, be, (short)0, accE, false, false);
            accA = __builtin_amdgcn_wmma_f32_16x16x4_f32(
                false, a, false, ba, (short)0, accA, false, false);
        }
#pragma unroll
        for (int j = 0; j < 8; ++j) {
            const int m   = j + ((lane >> 4) << 3);
            const int tok = tok0 + m;
            e_s[tok * DIM_S + n] = 1.0f / (1.0f + expf(-(accE[j] + eb[n])));
            a_s[tok * DIM_S + n] = tanhf(accA[j] + abv[n]);
        }
    }
}

// ---------------------------------------------------------------------------
// Kernel 2: serial scan, one block (64 threads = 2 waves) per batch element.
// Thread d owns Mv column d (50 VGPRs). w[t] is double-buffered in LDS:
// iteration t reads sw[t&1] while the next step's 50 values stream into
// sw[(t&1)^1] — via GLOBAL_LOAD_ASYNC_TO_LDS_B32 (ASYNCcnt) when available —
// giving exactly ONE barrier per serial step. Read uses the pre-update Mv.
// ---------------------------------------------------------------------------
__global__ __launch_bounds__(64) void k_scan(
    const float* __restrict__ Mv0,
    const float* __restrict__ w_s, const float* __restrict__ e_s,
    const float* __restrict__ a_s, float* __restrict__ rd_s)
{
    const int b = blockIdx.x;
    const int d = threadIdx.x;      // 0..63, column of Mv
    __shared__ float sw[2][64];     // double buffer (50 used per step)

    float mv[SIZE_M];
#pragma unroll
    for (int m = 0; m < SIZE_M; ++m) mv[m] = Mv0[m * DIM_S + d];

    const long base0 = (long)b * LL;

    // Preload step 0 into buffer 0.
#if HAVE_ASYNC_LDS
    if (d < SIZE_M)
        __builtin_amdgcn_global_load_async_to_lds_b32(
            (int*)(&w_s[base0 * SIZE_M + d]), (int*)(&sw[0][d]), 0, 0);
    __builtin_amdgcn_s_wait_asynccnt(0);
#else
    if (d < SIZE_M) sw[0][d] = w_s[base0 * SIZE_M + d];
#endif
    __syncthreads();

    for (int t = 0; t < LL; ++t) {
        const long base = base0 + t;
        const int  cur  = t & 1;
        const int  nxt  = cur ^ 1;

        // Kick off next step's w fetch into the other buffer (overlaps the
        // 50-FMA chain below; end-of-step barrier publishes it for t+1).
        if (d < SIZE_M && t + 1 < LL) {
#if HAVE_ASYNC_LDS
            __builtin_amdgcn_global_load_async_to_lds_b32(
                (int*)(&w_s[(base + 1) * SIZE_M + d]), (int*)(&sw[nxt][d]), 0, 0);
#else
            sw[nxt][d] = w_s[(base + 1) * SIZE_M + d];
#endif
        }
        if (t + 1 < LL) {
            __builtin_prefetch(&e_s[(base + 1) * DIM_S + d], 0, 0);  // global_prefetch_b8
            __builtin_prefetch(&a_s[(base + 1) * DIM_S + d], 0, 0);
        }

        const float et = e_s[base * DIM_S + d];
        const float at = a_s[base * DIM_S + d];
        float rd = 0.0f;
#pragma unroll
        for (int m = 0; m < SIZE_M; ++m) {
            const float wm = sw[cur][m];
            const float x  = mv[m];
            rd = fmaf(wm, x, rd);                        // read BEFORE update
            mv[m] = fmaf(wm, fmaf(-et, x, at), x);       // x*(1-wm*et)+wm*at
        }
        rd_s[base * DIM_S + d] = rd;

#if HAVE_ASYNC_LDS
        __builtin_amdgcn_s_wait_asynccnt(0);
#endif
        __syncthreads();   // single barrier per serial step
    }
}

// ---------------------------------------------------------------------------
// Kernel 3: per 16-token tile (one wave32).
//   f = tanh(concat(read,k) @ f_W + f_b)   (K=128, 32 WMMA k-steps per n-tile)
//   p = sigmoid(3*tanh(f@ab_W+ab_b) - tanh(k@d_W+d_b))
// ---------------------------------------------------------------------------
__global__ __launch_bounds__(32) void k_out(
    const float* __restrict__ rd_s, const float* __restrict__ k_s,
    const float* __restrict__ fW, const float* __restrict__ fb,
    const float* __restrict__ abW, const float* __restrict__ abb,
    const float* __restrict__ dW, const float* __restrict__ db,
    float* __restrict__ out)
{
    __shared__ float sA[16][130];   // cols 0..63 = read, 64..127 = k
    __shared__ float sF[16][65];

    const int lane = threadIdx.x;
    const int tok0 = blockIdx.x * 16;
    const int row  = lane & 15;
    const int koff = (lane >> 4) * 2;
    const int col0 = lane & 15;

    for (int i = 0; i < 16; ++i) {
        const int tok = tok0 + i;
        sA[i][lane]       = rd_s[tok * DIM_S + lane];
        sA[i][lane + 32]  = rd_s[tok * DIM_S + lane + 32];
        sA[i][64 + lane]  = k_s[tok * DIM_S + lane];
        sA[i][96 + lane]  = k_s[tok * DIM_S + lane + 32];
    }
    __syncthreads();

    for (int nt = 0; nt < 4; ++nt) {
        const int n = nt * 16 + col0;
        v8f acc = {};
#pragma unroll
        for (int kk = 0; kk < 32; ++kk) {
            const int kb = kk * 4 + koff;
            v2f a;  a.x = sA[row][kb];        a.y = sA[row][kb + 1];
            v2f bf; bf.x = fW[kb * DIM_S + n]; bf.y = fW[(kb + 1) * DIM_S + n];
            acc = __builtin_amdgcn_wmma_f32_16x16x4_f32(
                false, a, false, bf, (short)0, acc, false, false);
        }
#pragma unroll
        for (int j = 0; j < 8; ++j) {
            const int m = j + ((lane >> 4) << 3);
            sF[m][n] = tanhf(acc[j] + fb[n]);
        }
    }
    __syncthreads();

    if (lane < 16) {
        float sa = abb[0];
        float sd = db[0];
#pragma unroll
        for (int dd = 0; dd < DIM_S; ++dd) {
            sa += sF[lane][dd] * abW[dd];
            sd += sA[lane][64 + dd] * dW[dd];
        }
        const float abil = tanhf(sa);
        const float diff = tanhf(sd);
        out[tok0 + lane] = 1.0f / (1.0f + expf(-(3.0f * abil - diff)));
    }
}

// ---------------------------------------------------------------------------
extern "C" void kernel_launch(void* const* d_in, const int* in_sizes, int n_in,
                              void* d_out, int out_size, void* d_ws, size_t ws_size,
                              hipStream_t stream) {
    const int*   q     = (const int*)d_in[0];
    const int*   r     = (const int*)d_in[1];
    const float* k_emb = (const float*)d_in[2];
    const float* v_emb = (const float*)d_in[3];
    const float* Mk    = (const float*)d_in[4];
    const float* Mv0   = (const float*)d_in[5];
    const float* eW    = (const float*)d_in[6];
    const float* eb    = (const float*)d_in[7];
    const float* aW    = (const float*)d_in[8];
    const float* ab    = (const float*)d_in[9];
    const float* fW    = (const float*)d_in[10];
    const float* fb    = (const float*)d_in[11];
    const float* abW   = (const float*)d_in[12];
    const float* abb   = (const float*)d_in[13];
    const float* dW    = (const float*)d_in[14];
    const float* db    = (const float*)d_in[15];
    float* out = (float*)d_out;

    // Workspace layout (floats): k | w | e | a | reads  (~78 MB total)
    float* k_s  = (float*)d_ws;
    float* w_s  = k_s + (size_t)TOK * DIM_S;
    float* e_s  = w_s + (size_t)TOK * SIZE_M;
    float* a_s  = e_s + (size_t)TOK * DIM_S;
    float* rd_s = a_s + (size_t)TOK * DIM_S;

    k_gates<<<TOK / 16, 32, 0, stream>>>(q, r, k_emb, v_emb, Mk,
                                         eW, eb, aW, ab,
                                         k_s, w_s, e_s, a_s);
    k_scan<<<BB, 64, 0, stream>>>(Mv0, w_s, e_s, a_s, rd_s);
    k_out<<<TOK / 16, 32, 0, stream>>>(rd_s, k_s, fW, fb, abW, abb, dW, db, out);
}